// Clustering_87213605912783
// MI455X (gfx1250) — compile-verified
//
#include <hip/hip_runtime.h>
#include <math.h>

typedef __attribute__((ext_vector_type(2))) float v2f;
typedef __attribute__((ext_vector_type(8))) float v8f;

#define C_LANES 4            // instance clusters 1..4
#define EDIM    4            // embedding channels
#define BATCH   16
#define HW_N    (512*1024)   // pixels per image
#define BPB     128          // blocks per batch image
#define NTHREADS 256         // 8 waves (wave32)
#define NWAVES  8
#define PIX_BLOCK (HW_N / BPB)          // 4096 pixels / block
#define PIX_WAVE  (PIX_BLOCK / NWAVES)  // 512 pixels / wave
#define CHUNK   128                     // pixels per wave iteration
#define NCHUNK  (PIX_WAVE / CHUNK)      // 4
#define NSTAT   9            // 4 sums, 4 sumsq, 1 count
#define FSTR    132          // LDS row stride in dwords (132 mod 64 = 4)
#define LABROW  9
#define WAVE_F4 ((10*FSTR)/4)           // 330 float4 per wave region

__device__ __forceinline__ float4 f4mul(float4 a, float4 b) {
    float4 r; r.x=a.x*b.x; r.y=a.y*b.y; r.z=a.z*b.z; r.w=a.w*b.w; return r;
}

// ---------------------------------------------------------------------------
// Pass 1: WMMA one-hot segment reduction, 128 pixels / wave / chunk.
//   A (16x4 f32): one-hot of 4 pixels' labels (row m = label value)
//   B (4x16 f32): per-pixel feature rows [e0..e3, e0^2..e3^2, 1, dup...]
//   C/D (16x16 f32): running [sums | sumsq | count] per cluster row.
// LDS (per wave, stride 132 dwords): rows 0..3 = masked values,
// rows 4..7 = squares, row 8 = 1.0 (count, init once), row 9 = labels.
// Lanes nn>8 clamp to row 8 -> only D cols 9..15 (unused) see duplicates.
// ---------------------------------------------------------------------------
__global__ __launch_bounds__(NTHREADS)
void clu_accum(const float* __restrict__ pred,
               const float* __restrict__ bin,
               const int*   __restrict__ lab,
               float*       __restrict__ partials)
{
    __shared__ float4 stage4[NWAVES][WAVE_F4];           // 42240 B
    __shared__ float  red[NWAVES][C_LANES][NSTAT];       //  1152 B

    const int b    = blockIdx.y;
    const int wv   = threadIdx.x >> 5;
    const int lane = threadIdx.x & 31;
    const int nn   = lane & 15;      // B column n / A row m for this lane
    const int half = lane >> 4;      // 0: K slots {0,1}, 1: K slots {2,3}

    const float* pb = pred + (size_t)b * EDIM * HW_N;
    const float* bb = bin  + (size_t)b * HW_N;
    const int*   lb = lab  + (size_t)b * HW_N;

    const int wbase = blockIdx.x * PIX_BLOCK + wv * PIX_WAVE;
    float* my = (float*)stage4[wv];

    // one-time: count feature row = 1.0 for all 128 pixel columns
    {
        float4 one; one.x = one.y = one.z = one.w = 1.f;
        *(float4*)&my[8*FSTR + 4*lane] = one;
    }
    __builtin_amdgcn_wave_barrier();

    const int   bbase = ((nn < NSTAT) ? nn : 8) * FSTR + 2*half;
    const int   lbase = LABROW * FSTR + 2*half;
    const float fm    = (float)nn;

    v8f acc = {0.f,0.f,0.f,0.f,0.f,0.f,0.f,0.f};

    for (int it = 0; it < NCHUNK; ++it) {
        const int idx4 = wbase + it*CHUNK + 4*lane;

        // prefetch next chunk: 4 lanes x 6 streams cover all 24 cachelines
        if (lane < 4 && it + 1 < NCHUNK) {
            const int nx = wbase + (it+1)*CHUNK + 32*lane;
            __builtin_prefetch(&pb[0*HW_N + nx], 0, 1);
            __builtin_prefetch(&pb[1*HW_N + nx], 0, 1);
            __builtin_prefetch(&pb[2*HW_N + nx], 0, 1);
            __builtin_prefetch(&pb[3*HW_N + nx], 0, 1);
            __builtin_prefetch(&bb[nx], 0, 1);
            __builtin_prefetch(&lb[nx], 0, 1);
        }

        // 6 x global_load_b128: 24 B / pixel total, fully coalesced
        const float4 w4 = *(const float4*)(bb + idx4);
        const int4   l4 = *(const int4*)  (lb + idx4);
        float4 v0 = *(const float4*)(pb + 0*HW_N + idx4);
        float4 v1 = *(const float4*)(pb + 1*HW_N + idx4);
        float4 v2 = *(const float4*)(pb + 2*HW_N + idx4);
        float4 v3 = *(const float4*)(pb + 3*HW_N + idx4);
        v0 = f4mul(v0, w4);  v1 = f4mul(v1, w4);
        v2 = f4mul(v2, w4);  v3 = f4mul(v3, w4);
        const float4 s0 = f4mul(v0, v0), s1 = f4mul(v1, v1);
        const float4 s2 = f4mul(v2, v2), s3 = f4mul(v3, v3);
        float4 lf; lf.x=(float)l4.x; lf.y=(float)l4.y;
                   lf.z=(float)l4.z; lf.w=(float)l4.w;

        // 9 x ds_store_b128 (channel-major staging)
        const int col = 4*lane;
        *(float4*)&my[0*FSTR + col] = v0;
        *(float4*)&my[1*FSTR + col] = v1;
        *(float4*)&my[2*FSTR + col] = v2;
        *(float4*)&my[3*FSTR + col] = v3;
        *(float4*)&my[4*FSTR + col] = s0;
        *(float4*)&my[5*FSTR + col] = s1;
        *(float4*)&my[6*FSTR + col] = s2;
        *(float4*)&my[7*FSTR + col] = s3;
        *(float4*)&my[LABROW*FSTR + col] = lf;
        __builtin_amdgcn_wave_barrier();

        // 32 WMMAs of 4 pixels each; branch-free operand builds
#pragma unroll 8
        for (int k0 = 0; k0 < CHUNK; k0 += 4) {
            const float b0  = my[bbase + k0];        // ds_load_b64 pair
            const float b1  = my[bbase + k0 + 1];
            const float l02 = my[lbase + k0];        // ds_load_b64 pair
            const float l13 = my[lbase + k0 + 1];
            v2f A;  A.x = (l02 == fm) ? 1.f : 0.f;
                    A.y = (l13 == fm) ? 1.f : 0.f;
            v2f Bv; Bv.x = b0; Bv.y = b1;
            acc = __builtin_amdgcn_wmma_f32_16x16x4_f32(
                      false, A, false, Bv, (short)0, acc, false, false);
        }
        __builtin_amdgcn_wave_barrier();
    }

    // D rows m=1..4 live in acc[1..4] of lanes 0..15 (cols n=0..8)
    if (half == 0 && nn < NSTAT) {
        red[wv][0][nn] = acc[1];
        red[wv][1][nn] = acc[2];
        red[wv][2][nn] = acc[3];
        red[wv][3][nn] = acc[4];
    }
    __syncthreads();

    // deterministic in-block reduction over the 8 waves
    const int t = threadIdx.x;
    if (t < C_LANES * NSTAT) {
        const int c = t / NSTAT, s = t % NSTAT;
        float v = 0.f;
#pragma unroll
        for (int wvi = 0; wvi < NWAVES; ++wvi) v += red[wvi][c][s];
        partials[(((size_t)b * BPB + blockIdx.x) * C_LANES + c) * NSTAT + s] = v;
    }
}

// ---------------------------------------------------------------------------
// Pass 2: deterministic reduction of block partials + loss tail.
// ---------------------------------------------------------------------------
__global__ __launch_bounds__(64)
void clu_final(const float* __restrict__ partials, float* __restrict__ out)
{
    __shared__ float mu[BATCH][C_LANES][EDIM];
    __shared__ float vterm[BATCH][C_LANES];
    const int t = threadIdx.x;

    if (t < BATCH * C_LANES) {
        const int b = t >> 2, c = t & 3;
        float s[NSTAT];
#pragma unroll
        for (int k = 0; k < NSTAT; ++k) s[k] = 0.f;
        for (int blk = 0; blk < BPB; ++blk) {
            const float* p = partials + (((size_t)b*BPB + blk)*C_LANES + c)*NSTAT;
#pragma unroll
            for (int k = 0; k < NSTAT; ++k) s[k] += p[k];
        }
        const float cnt = s[8];
        const float inv = 1.f / cnt;
        const float m0 = s[0]*inv, m1 = s[1]*inv, m2 = s[2]*inv, m3 = s[3]*inv;
        mu[b][c][0]=m0; mu[b][c][1]=m1; mu[b][c][2]=m2; mu[b][c][3]=m3;
        // ssd = sumsq - ||sum||^2 / count
        float ssd = (s[4]+s[5]+s[6]+s[7])
                  - (s[0]*s[0] + s[1]*s[1] + s[2]*s[2] + s[3]*s[3]) * inv;
        ssd = fmaxf(ssd, 0.f);
        const float nrm = sqrtf(ssd);
        const float dv  = nrm - 0.5f;                 // DELTA_V
        vterm[b][c] = (nrm > 0.5f) ? dv*dv : 0.f;
    }
    __syncthreads();

    if (t == 0) {
        float lvar = 0.f;
        for (int b = 0; b < BATCH; ++b)
            for (int c = 0; c < C_LANES; ++c) lvar += vterm[b][c];
        lvar *= (1.f / (BATCH * C_LANES));

        float ldist = 0.f;
        for (int b = 0; b < BATCH; ++b) {
            for (int i = 0; i < C_LANES; ++i) {
                for (int j = 0; j < C_LANES; ++j) {
                    if (i == j) continue;
                    const float d0 = mu[b][i][0] - mu[b][j][0];
                    const float d1 = mu[b][i][1] - mu[b][j][1];
                    const float d2 = mu[b][i][2] - mu[b][j][2];
                    const float d3 = mu[b][i][3] - mu[b][j][3];
                    const float dd = sqrtf(d0*d0 + d1*d1 + d2*d2 + d3*d3);
                    const float r  = 3.0f - dd;       // DELTA_D
                    if (r > 0.f) ldist += r * r;
                }
            }
        }
        ldist *= (1.f / BATCH);
        out[0] = lvar + ldist;
    }
}

// ---------------------------------------------------------------------------
extern "C" void kernel_launch(void* const* d_in, const int* in_sizes, int n_in,
                              void* d_out, int out_size, void* d_ws, size_t ws_size,
                              hipStream_t stream)
{
    const float* pred = (const float*)d_in[0];
    const float* bin  = (const float*)d_in[1];
    const int*   lab  = (const int*)d_in[2];
    float* out      = (float*)d_out;
    float* partials = (float*)d_ws;   // BATCH*BPB*C_LANES*NSTAT f32 = 288 KiB

    dim3 grid(BPB, BATCH, 1);
    clu_accum<<<grid, NTHREADS, 0, stream>>>(pred, bin, lab, partials);
    clu_final<<<1, 64, 0, stream>>>(partials, out);

    (void)in_sizes; (void)n_in; (void)out_size; (void)ws_size;
}